// SelectiveSSM_4011499454827
// MI455X (gfx1250) — compile-verified
//
#include <hip/hip_runtime.h>
#include <hip/hip_bf16.h>

// ---------------------------------------------------------------------------
// Selective SSM (Mamba-style) forward for MI455X / gfx1250 (wave32, WMMA).
//   B=2, L=2048, D_MODEL=1024, D_STATE=16, DT_RANK=64
// Two fp32 WMMA GEMM kernels + one LDS-staged sequential scan kernel.
// ---------------------------------------------------------------------------

#define BATCH   2
#define SEQ     2048
#define DMODEL  1024
#define NSTATE  16
#define DTRANK  64
#define MTOT    (BATCH * SEQ)      // 4096 flattened rows
#define TCH     64                 // scan LDS chunk (timesteps)

typedef float v2f __attribute__((ext_vector_type(2)));
typedef float v8f __attribute__((ext_vector_type(8)));

// ---------------------------------------------------------------------------
// Kernel 1: fused projection  xp = x * [W_x ; W_C]^T   (N = 64 + 16 + 16 = 96)
// One wave (32 threads) computes a 16x16 tile; K-loop over 1024 in steps of 4
// with V_WMMA_F32_16X16X4_F32 (full fp32 accuracy).
// Outputs routed: n<64 -> delta_low, 64<=n<80 -> Bmat, n>=80 -> Cmat.
// ---------------------------------------------------------------------------
__global__ void ssm_proj_kernel(const float* __restrict__ x,
                                const float* __restrict__ Wx,   // [80][1024]
                                const float* __restrict__ Wc,   // [16][1024]
                                float* __restrict__ dl,         // [4096][64]
                                float* __restrict__ Bo,         // [4096][16]
                                float* __restrict__ Co)         // [4096][16]
{
    const int m0   = blockIdx.x * 16;         // row tile (flattened b*L)
    const int n0   = blockIdx.y * 16;         // output-column tile (0..95)
    const int lane = threadIdx.x;             // 0..31 (wave32)
    const int half = lane >> 4;               // K split: 0 -> K{0,1}, 1 -> K{2,3}
    const int row  = lane & 15;

    const int gn = n0 + row;                  // global output column for B operand
    const float* wrow = (gn < 80) ? (Wx + (size_t)gn * DMODEL)
                                  : (Wc + (size_t)(gn - 80) * DMODEL);
    const float* arow = x + (size_t)(m0 + row) * DMODEL + 2 * half;
    const float* brow = wrow + 2 * half;

    v8f c = {};
    for (int k = 0; k < DMODEL; k += 4) {
        v2f a, b;
        a.x = arow[k];     a.y = arow[k + 1];
        b.x = brow[k];     b.y = brow[k + 1];
        c = __builtin_amdgcn_wmma_f32_16x16x4_f32(
                /*neg_a=*/false, a, /*neg_b=*/false, b,
                /*c_mod=*/(short)0, c, /*reuse_a=*/false, /*reuse_b=*/false);
    }

    // D layout: VGPR v, lanes 0-15: (M=v, N=lane); lanes 16-31: (M=v+8, N=lane-16)
#pragma unroll
    for (int v = 0; v < 8; ++v) {
        const int mg  = m0 + v + 8 * half;
        const float val = c[v];
        if (gn < 64)      dl[(size_t)mg * DTRANK + gn]        = val;
        else if (gn < 80) Bo[(size_t)mg * NSTATE + (gn - 64)] = val;
        else              Co[(size_t)mg * NSTATE + (gn - 80)] = val;
    }
}

// ---------------------------------------------------------------------------
// Kernel 2: delta = softplus(delta_low * W_dt^T + b_dt)   [4096 x 1024]
// K = 64, so 16 WMMAs per tile; fused bias + stable softplus epilogue.
// ---------------------------------------------------------------------------
__global__ void ssm_delta_kernel(const float* __restrict__ dl,   // [4096][64]
                                 const float* __restrict__ Wdt,  // [1024][64]
                                 const float* __restrict__ bdt,  // [1024]
                                 float* __restrict__ delta)      // [4096][1024]
{
    const int m0   = blockIdx.x * 16;
    const int n0   = blockIdx.y * 16;
    const int lane = threadIdx.x;
    const int half = lane >> 4;
    const int row  = lane & 15;

    const int gn = n0 + row;
    const float* arow = dl  + (size_t)(m0 + row) * DTRANK + 2 * half;
    const float* brow = Wdt + (size_t)gn * DTRANK + 2 * half;  // B[k][n] = Wdt[n][k]

    v8f c = {};
#pragma unroll
    for (int k = 0; k < DTRANK; k += 4) {
        v2f a, b;
        a.x = arow[k];     a.y = arow[k + 1];
        b.x = brow[k];     b.y = brow[k + 1];
        c = __builtin_amdgcn_wmma_f32_16x16x4_f32(
                false, a, false, b, (short)0, c, false, false);
    }

    const float bias = bdt[gn];
#pragma unroll
    for (int v = 0; v < 8; ++v) {
        const int mg = m0 + v + 8 * half;
        const float z  = c[v] + bias;
        // numerically stable softplus
        const float sp = (z > 20.0f) ? z : log1pf(__expf(z));
        delta[(size_t)mg * DMODEL + gn] = sp;
    }
}

// ---------------------------------------------------------------------------
// Kernel 3: sequential selective scan over L.
// Thread (b, d) owns h[0..15] in VGPRs; per-timestep broadcast vectors
// B_t/C_t are chunk-staged into LDS (TCH steps = 8 KB).
//   h[n] = exp(delta*A[d][n]) * h[n] + (delta * B_t[n]) * x_t
//   y    = sum_n C_t[n] * h[n]
// ---------------------------------------------------------------------------
__global__ void ssm_scan_kernel(const float* __restrict__ x,      // [B][L][D]
                                const float* __restrict__ A_log,  // [D][16]
                                const float* __restrict__ delta,  // [B*L][D]
                                const float* __restrict__ Bw,     // [B*L][16]
                                const float* __restrict__ Cw,     // [B*L][16]
                                float* __restrict__ out)          // [B][L][D]
{
    const int d   = blockIdx.x * blockDim.x + threadIdx.x;  // 0..1023
    const int b   = blockIdx.y;
    const int tid = threadIdx.x;

    __shared__ float sB[TCH * NSTATE];
    __shared__ float sC[TCH * NSTATE];

    float Ar[NSTATE], h[NSTATE];
#pragma unroll
    for (int n = 0; n < NSTATE; ++n) {
        Ar[n] = -__expf(A_log[(size_t)d * NSTATE + n]);   // A = -exp(A_log)
        h[n]  = 0.0f;
    }

    const size_t rowbase = (size_t)b * SEQ;
    const float* dptr = delta + rowbase * DMODEL + d;
    const float* xptr = x     + rowbase * DMODEL + d;
    float*       optr = out   + rowbase * DMODEL + d;
    const float* bptr = Bw + rowbase * NSTATE;
    const float* cptr = Cw + rowbase * NSTATE;

    for (int t0 = 0; t0 < SEQ; t0 += TCH) {
        __syncthreads();
        for (int i = tid; i < TCH * NSTATE; i += blockDim.x) {
            sB[i] = bptr[(size_t)t0 * NSTATE + i];
            sC[i] = cptr[(size_t)t0 * NSTATE + i];
        }
        __syncthreads();

        for (int tt = 0; tt < TCH; ++tt) {
            const int t = t0 + tt;
            const float dlt = dptr[(size_t)t * DMODEL];
            const float xt  = xptr[(size_t)t * DMODEL];
            float y = 0.0f;
#pragma unroll
            for (int n = 0; n < NSTATE; ++n) {
                const float ab = __expf(dlt * Ar[n]);
                h[n] = ab * h[n] + (dlt * sB[tt * NSTATE + n]) * xt;
                y += sC[tt * NSTATE + n] * h[n];
            }
            optr[(size_t)t * DMODEL] = y;
        }
    }
}

// ---------------------------------------------------------------------------
// Host launch
// ---------------------------------------------------------------------------
extern "C" void kernel_launch(void* const* d_in, const int* in_sizes, int n_in,
                              void* d_out, int out_size, void* d_ws, size_t ws_size,
                              hipStream_t stream)
{
    (void)in_sizes; (void)n_in; (void)out_size; (void)ws_size;

    const float* x     = (const float*)d_in[0];  // (2, 2048, 1024)
    const float* A_log = (const float*)d_in[1];  // (1024, 16)
    const float* W_x   = (const float*)d_in[2];  // (80, 1024)
    const float* W_dt  = (const float*)d_in[3];  // (1024, 64)
    const float* b_dt  = (const float*)d_in[4];  // (1024,)
    const float* W_C   = (const float*)d_in[5];  // (16, 1024)
    float* out = (float*)d_out;                  // (2, 2048, 1024)

    // Workspace layout (floats): delta | delta_low | Bmat | Cmat  (~18.4 MB)
    float* ws      = (float*)d_ws;
    float* wsDelta = ws;                                   // MTOT * DMODEL
    float* wsDL    = wsDelta + (size_t)MTOT * DMODEL;      // MTOT * DTRANK
    float* wsB     = wsDL    + (size_t)MTOT * DTRANK;      // MTOT * NSTATE
    float* wsC     = wsB     + (size_t)MTOT * NSTATE;      // MTOT * NSTATE

    // 1) fused x-projection: 96 output cols (delta_low | B | C)
    ssm_proj_kernel<<<dim3(MTOT / 16, 96 / 16), 32, 0, stream>>>(
        x, W_x, W_C, wsDL, wsB, wsC);

    // 2) delta GEMM + bias + softplus
    ssm_delta_kernel<<<dim3(MTOT / 16, DMODEL / 16), 32, 0, stream>>>(
        wsDL, W_dt, b_dt, wsDelta);

    // 3) sequential scan (parallel over b x d)
    ssm_scan_kernel<<<dim3(DMODEL / 256, BATCH), 256, 0, stream>>>(
        x, A_log, wsDelta, wsB, wsC, out);
}